// SwappingCorruption_7438883357300
// MI455X (gfx1250) — compile-verified
//
#include <hip/hip_runtime.h>
#include <stdint.h>

// SwappingCorruption: out[i,j] = swap_mask[i,j] ? x[i, perm[i,j]] : x[i,j]
//
// MI455X / gfx1250 strategy:
//   - One workgroup per row (8 KiB f32 row << 320 KB LDS/WGP).
//   - Stage the row into LDS with gfx1250 async global->LDS B128 copies
//     (ASYNCcnt path): the random intra-row gather is then served by the
//     64-bank LDS instead of random vmem addresses.
//   - All global traffic (x, perm, mask in; out) is coalesced B128/B32
//     streaming -> HBM roofline (~436 MB total, ~19 us at 23.3 TB/s).
//     Mask sparsity cannot reduce perm traffic: P(128B line touched) =
//     1 - 0.9^32 ~= 97%, so dense vectorized loads are optimal.

#define N_FEAT 2048
#define BLOCK  256

__global__ __launch_bounds__(BLOCK) void swapping_corruption_kernel(
    const float* __restrict__ x,
    const unsigned char* __restrict__ mask,   // JAX bool -> 1 byte/elem
    const int* __restrict__ perm,
    float* __restrict__ out)
{
    __shared__ float row[N_FEAT];   // 8192 B

    const int tid = threadIdx.x;
    const size_t rbase = (size_t)blockIdx.x * N_FEAT;

    const float*         xrow = (const float*)__builtin_assume_aligned(x + rbase, 16);
    const unsigned char* mrow = (const unsigned char*)__builtin_assume_aligned(mask + rbase, 4);
    const int*           prow = (const int*)__builtin_assume_aligned(perm + rbase, 16);
    float*               orow = (float*)__builtin_assume_aligned(out + rbase, 16);

    // ---- Phase 1: async-stage the row into LDS (2 x B128 per lane) ----
#pragma unroll
    for (int p = 0; p < 2; ++p) {
        const int e = p * (BLOCK * 4) + tid * 4;          // float index
        // addrspace(3) offset == low 32 bits of the generic shared address
        const uint32_t lds_off = (uint32_t)(uintptr_t)(&row[e]);
        const float* ga = xrow + e;
        asm volatile("global_load_async_to_lds_b128 %0, %1, off"
                     :: "v"(lds_off), "v"(ga)
                     : "memory");
    }
    // Per-wave: drain our async copies, then workgroup barrier so every
    // wave sees the fully-populated row.
    asm volatile("s_wait_asynccnt 0" ::: "memory");
    __syncthreads();

    // ---- Phase 2: branchless gather/select out of LDS, B128 store ----
#pragma unroll
    for (int p = 0; p < 2; ++p) {
        const int j = p * (BLOCK * 4) + tid * 4;

        const int4   pv = *(const int4*)__builtin_assume_aligned(prow + j, 16);
        const uchar4 mv = *(const uchar4*)__builtin_assume_aligned(mrow + j, 4);

        float4 o;
        o.x = mv.x ? row[pv.x] : row[j + 0];
        o.y = mv.y ? row[pv.y] : row[j + 1];
        o.z = mv.z ? row[pv.z] : row[j + 2];
        o.w = mv.w ? row[pv.w] : row[j + 3];

        *(float4*)__builtin_assume_aligned(orow + j, 16) = o;
    }
}

extern "C" void kernel_launch(void* const* d_in, const int* in_sizes, int n_in,
                              void* d_out, int out_size, void* d_ws, size_t ws_size,
                              hipStream_t stream) {
    (void)n_in; (void)out_size; (void)d_ws; (void)ws_size;

    const float*         x    = (const float*)d_in[0];
    const unsigned char* mask = (const unsigned char*)d_in[1];  // bool (1B)
    const int*           perm = (const int*)d_in[2];
    float*               out  = (float*)d_out;

    const int rows = in_sizes[0] / N_FEAT;   // 16384

    swapping_corruption_kernel<<<rows, BLOCK, 0, stream>>>(x, mask, perm, out);
}